// _GraphConv_68143951118763
// MI455X (gfx1250) — compile-verified
//
#include <hip/hip_runtime.h>
#include <hip/hip_bf16.h>

// Problem constants (from reference)
#define J_   17
#define C_   16
#define JC_  (J_ * C_)    // 272
#define N_   32
#define H_   64
#define W_   64
#define HW_  4096
#define KK_  (C_ * 9)     // 144
#define WSTR 146          // padded LDS row stride for weights (even, ~conflict-free)
#define BN_EPS_ 1e-5f

typedef __attribute__((ext_vector_type(2))) float v2f;
typedef __attribute__((ext_vector_type(8))) float v8f;

// LDS element offset of im2col K-index k relative to &xt[0][r][col]
// xt layout: [ci][4 rows][66 cols] -> ci stride 264, row stride 66, col stride 1
__host__ __device__ constexpr int xoff(int k) {
    return (k / 9) * (4 * 66) + ((k % 9) / 3) * 66 + (k % 3);
}

// ---------------------------------------------------------------------------
// K0: row softmax of 17x17 adjacency (one tiny block)
// ---------------------------------------------------------------------------
__global__ void k0_softmax(const float* __restrict__ adj, float* __restrict__ adjsm) {
    int i = threadIdx.x;
    if (i < J_) {
        float m = -1e30f;
        for (int j = 0; j < J_; ++j) m = fmaxf(m, adj[i * J_ + j]);
        float e[J_];
        float s = 0.f;
        for (int j = 0; j < J_; ++j) { e[j] = expf(adj[i * J_ + j] - m); s += e[j]; }
        float inv = 1.f / s;
        for (int j = 0; j < J_; ++j) adjsm[i * J_ + j] = e[j] * inv;
    }
}

// ---------------------------------------------------------------------------
// K1: grouped 3x3 conv as implicit GEMM on V_WMMA_F32_16X16X4_F32.
// Per block: one (n, j) group, a 2-row x 64-col strip (128 pixels).
// 8 waves; each wave owns a 16(co) x 16(pixel) WMMA tile; K = 144 = 36 steps,
// fully unrolled so LDS addresses are base+imm (one cndmask per B operand for
// the lane-half K-select). Epilogue fuses bias + A1/A2 reductions.
// ---------------------------------------------------------------------------
__global__ __launch_bounds__(256)
void k1_conv(const float* __restrict__ x, const float* __restrict__ Wc,
             const float* __restrict__ Wb, const float* __restrict__ att_w,
             float* __restrict__ Bout, float* __restrict__ A1out,
             float* __restrict__ A2out) {
    __shared__ float wts[C_][WSTR];    // 16 x 144 group weights, padded stride
    __shared__ float xt[C_][4][66];    // ci x 4 rows x 66 cols (1-col halo each side)
    __shared__ float wbs[C_];
    __shared__ float aw[2 * C_];

    const int bid   = blockIdx.x;
    const int strip = bid & 31;        // 32 strips of 2 rows
    const int nj    = bid >> 5;
    const int n     = nj / J_;
    const int j     = nj % J_;
    const int tid   = threadIdx.x;
    const int lane  = tid & 31;
    const int wv    = tid >> 5;
    const int h0    = strip * 2;

    // Stage group weights: Wc[(j*16+co)*144 + k] is already [co][k] row-major.
    const float* wg = Wc + (size_t)(j * C_) * KK_;
    for (int idx = tid; idx < C_ * KK_; idx += 256)
        wts[idx / KK_][idx % KK_] = wg[idx];
    if (tid < C_)     wbs[tid] = Wb[j * C_ + tid];
    if (tid < 2*C_)   aw[tid]  = att_w[tid];

    // Stage haloed x strip: rows h0-1 .. h0+2, cols -1 .. 64, zero padded.
    const float* xg = x + ((size_t)n * JC_ + (size_t)j * C_) * HW_;
    for (int idx = tid; idx < C_ * 4 * 66; idx += 256) {
        int ci  = idx / (4 * 66);
        int rem = idx % (4 * 66);
        int dr  = rem / 66;
        int cc  = rem % 66 - 1;
        int row = h0 - 1 + dr;
        float v = 0.f;
        if (row >= 0 && row < H_ && cc >= 0 && cc < W_)
            v = xg[(size_t)ci * HW_ + row * W_ + cc];
        xt[ci][dr][rem % 66] = v;
    }
    __syncthreads();

    // Wave tile: pixel q = wv*16 + (lane&15) within the 128-pixel strip.
    const int pl   = lane & 15;
    const int q    = wv * 16 + pl;
    const int r    = q >> 6;        // local row 0/1
    const int col  = q & 63;
    const int half = lane >> 4;     // K-pair select per ISA A/B layout

    v8f acc = {};
#if defined(__HIP_DEVICE_COMPILE__) && __has_builtin(__builtin_amdgcn_wmma_f32_16x16x4_f32)
    // A 16x4 layout: lanes 0-15 -> M=lane, K={k0,k0+1}; lanes 16-31 -> K={k0+2,k0+3}.
    // B 4x16 layout assumed symmetric: N=lane&15, same K split.
    const float* pa = &wts[pl][half * 2];        // 8B aligned (WSTR even)
    const float* pb = &xt[0][r][col];
#pragma unroll
    for (int s = 0; s < 36; ++s) {
        v2f a = *(const v2f*)(pa + s * 4);       // ds_load_b64 @ imm offset
        v2f b;
        const int e0 = half ? xoff(s * 4 + 2) : xoff(s * 4);
        const int e1 = half ? xoff(s * 4 + 3) : xoff(s * 4 + 1);
        b.x = pb[e0];
        b.y = pb[e1];
        // 8 args: (neg_a, A, neg_b, B, c_mod, C, reuse_a, reuse_b)
        acc = __builtin_amdgcn_wmma_f32_16x16x4_f32(
            false, a, false, b, (short)0, acc, false, false);
    }
#else
    // Direct-FMA fallback with identical C/D lane layout (host parse / no builtin).
    for (int s = 0; s < 36; ++s) {
        for (int rr = 0; rr < 8; ++rr) {
            int co = rr + half * 8;
            float sum = acc[rr];
            for (int dk = 0; dk < 4; ++dk) {
                int kk = s * 4 + dk;
                int ci = kk / 9, t9 = kk % 9, kh = t9 / 3, kw = t9 % 3;
                sum = fmaf(wts[co][kk], xt[ci][r + kh][col + kw], sum);
            }
            acc[rr] = sum;
        }
    }
#endif

    // Epilogue: bias, store B, fold A1/A2 channel dot-products.
    // C/D layout: VGPR rr, lane -> co = rr + 8*half, pixel = lane&15.
    const size_t outbase = ((size_t)n * JC_ + (size_t)j * C_) * HW_ + (size_t)h0 * W_;
    float p1 = 0.f, p2 = 0.f;
#pragma unroll
    for (int rr = 0; rr < 8; ++rr) {
        int co = rr + half * 8;
        float v = acc[rr] + wbs[co];
        Bout[outbase + (size_t)co * HW_ + q] = v;
        p1 += v * aw[co];
        p2 += v * aw[C_ + co];
    }
    // Sum the two lane-halves (co 0-7 and 8-15 live in lanes L and L^16).
    float f1 = p1 + __shfl_xor(p1, 16, 32);
    float f2 = p2 + __shfl_xor(p2, 16, 32);
    if (half == 0) {
        size_t abase = ((size_t)n * J_ + j) * HW_ + (size_t)h0 * W_ + q;
        A1out[abase] = f1;
        A2out[abase] = f2;
    }
}

// ---------------------------------------------------------------------------
// K2: attention mix. Block = (n, 16-pixel tile). Hoist the attention weight
// w[i][j][p] = adj_sm[i][j]*relu(A1[i][p]+A2[j][p]+b) into LDS once (kills the
// 16x redundancy over c), then each thread (c,p) accumulates over j for all i.
// Writes z into d_out (normalized in-place later).
// ---------------------------------------------------------------------------
__global__ __launch_bounds__(256)
void k2_mix(const float* __restrict__ Bbuf, const float* __restrict__ A1in,
            const float* __restrict__ A2in, const float* __restrict__ adjsm,
            const float* __restrict__ att_b, float* __restrict__ z) {
    __shared__ float BrT[J_][C_][16];    // j, c, p
    __shared__ float wj[J_][J_][16];     // i, j, p
    __shared__ float a1s[J_][16];
    __shared__ float a2s[J_][16];
    __shared__ float adj_s[J_ * J_];

    const int bid  = blockIdx.x;
    const int n    = bid >> 8;           // 256 pixel tiles per image
    const int p0   = (bid & 255) * 16;
    const int tid  = threadIdx.x;

    for (int idx = tid; idx < J_ * C_ * 16; idx += 256) {
        int jj = idx / (C_ * 16);
        int rem = idx % (C_ * 16);
        int c = rem >> 4, p = rem & 15;
        BrT[jj][c][p] = Bbuf[((size_t)n * JC_ + jj * C_ + c) * HW_ + p0 + p];
    }
    for (int idx = tid; idx < J_ * 16; idx += 256) {
        int jj = idx >> 4, p = idx & 15;
        size_t base = ((size_t)n * J_ + jj) * HW_ + p0 + p;
        a1s[jj][p] = A1in[base];
        a2s[jj][p] = A2in[base];
    }
    for (int idx = tid; idx < J_ * J_; idx += 256) adj_s[idx] = adjsm[idx];
    __syncthreads();

    const float ab = att_b[0];
    for (int idx = tid; idx < J_ * J_ * 16; idx += 256) {
        int i = idx / (J_ * 16);
        int rem = idx % (J_ * 16);
        int jj = rem >> 4, p = rem & 15;
        float s = fmaxf(a1s[i][p] + a2s[jj][p] + ab, 0.f);
        wj[i][jj][p] = adj_s[i * J_ + jj] * s;
    }
    __syncthreads();

    const int c = tid & 15;
    const int p = tid >> 4;
    for (int i = 0; i < J_; ++i) {
        float acc = 0.f;
#pragma unroll
        for (int jj = 0; jj < J_; ++jj)
            acc += wj[i][jj][p] * BrT[jj][c][p];
        z[((size_t)n * JC_ + i * C_ + c) * HW_ + p0 + p] = acc;
    }
}

// ---------------------------------------------------------------------------
// K3: per-channel BN statistics (mean, rsqrt(var+eps)). One block per channel.
// ---------------------------------------------------------------------------
__global__ __launch_bounds__(256)
void k3_stats(const float* __restrict__ z, float* __restrict__ mean,
              float* __restrict__ rstd) {
    const int ch  = blockIdx.x;    // 0..271
    const int tid = threadIdx.x;
    float s = 0.f, s2 = 0.f;
    for (int n = 0; n < N_; ++n) {
        const float* zp = z + ((size_t)n * JC_ + ch) * HW_;
        for (int p = tid; p < HW_; p += 256) {
            float v = zp[p];
            s += v;
            s2 += v * v;
        }
    }
    __shared__ float rs[256], rs2[256];
    rs[tid] = s; rs2[tid] = s2;
    __syncthreads();
    for (int st = 128; st > 0; st >>= 1) {
        if (tid < st) { rs[tid] += rs[tid + st]; rs2[tid] += rs2[tid + st]; }
        __syncthreads();
    }
    if (tid == 0) {
        const float cnt = (float)(N_ * HW_);
        float m = rs[0] / cnt;
        float var = rs2[0] / cnt - m * m;
        mean[ch] = m;
        rstd[ch] = rsqrtf(var + BN_EPS_);
    }
}

// ---------------------------------------------------------------------------
// K4: in-place normalize + affine + relu on d_out.
// ---------------------------------------------------------------------------
__global__ __launch_bounds__(256)
void k4_bn(float* __restrict__ out, const float* __restrict__ mean,
           const float* __restrict__ rstd, const float* __restrict__ gamma,
           const float* __restrict__ beta) {
    size_t idx = (size_t)blockIdx.x * 256 + threadIdx.x;
    const size_t total = (size_t)N_ * JC_ * HW_;
    if (idx < total) {
        int ch = (int)((idx / HW_) % JC_);
        float v = out[idx];
        v = (v - mean[ch]) * rstd[ch] * gamma[ch] + beta[ch];
        out[idx] = fmaxf(v, 0.f);
    }
}

// ---------------------------------------------------------------------------
// Launch
// ---------------------------------------------------------------------------
extern "C" void kernel_launch(void* const* d_in, const int* in_sizes, int n_in,
                              void* d_out, int out_size, void* d_ws, size_t ws_size,
                              hipStream_t stream) {
    const float* x      = (const float*)d_in[0];
    const float* adj    = (const float*)d_in[1];
    const float* Wc     = (const float*)d_in[2];
    const float* Wb     = (const float*)d_in[3];
    const float* att_w  = (const float*)d_in[4];
    const float* att_b  = (const float*)d_in[5];
    const float* gamma  = (const float*)d_in[6];
    const float* beta   = (const float*)d_in[7];
    float* out = (float*)d_out;

    // Workspace layout (floats)
    float* ws = (float*)d_ws;
    const size_t sz_a = (size_t)N_ * J_ * HW_;       // 2,228,224
    const size_t sz_b = (size_t)N_ * JC_ * HW_;      // 35,651,584
    float* adjsm = ws;                 // 512 slot
    float* A1    = ws + 512;
    float* A2    = A1 + sz_a;
    float* Bbuf  = A2 + sz_a;
    float* meanb = Bbuf + sz_b;
    float* rstdb = meanb + 512;

    k0_softmax<<<1, 32, 0, stream>>>(adj, adjsm);
    k1_conv<<<N_ * J_ * 32, 256, 0, stream>>>(x, Wc, Wb, att_w, Bbuf, A1, A2);
    k2_mix<<<N_ * 256, 256, 0, stream>>>(Bbuf, A1, A2, adjsm, att_b, out);
    k3_stats<<<JC_, 256, 0, stream>>>(out, meanb, rstdb);
    const size_t total = (size_t)N_ * JC_ * HW_;
    k4_bn<<<(unsigned)((total + 255) / 256), 256, 0, stream>>>(out, meanb, rstdb, gamma, beta);
    (void)in_sizes; (void)n_in; (void)out_size; (void)ws_size;
}